// CLIPAttentionPooling_8426725835102
// MI455X (gfx1250) — compile-verified
//
#include <hip/hip_runtime.h>

// CLIP attention pooling for MI455X (gfx1250), fp32 WMMA path.
//
// Roofline: 86 GFLOP total; full working set (inputs 25MB + Q/K/S scratch
// 96MB) fits in 192MB L2, so HBM time ~2us at 23.3 TB/s -> matrix-pipe bound.
// fp32 WMMA (V_WMMA_F32_16X16X4_F32) is mandatory: logits have std~32 and
// softmax exponentiates them, so bf16's ~2^-8 relative input error would turn
// into O(1) attention-weight error after exp().
//
// Register blocking: one wave owns a 32x64 output strip (8 f32 accumulator
// tiles = 64 VGPRs). Per K-step of 4: 2 A-fragment loads + 4 B-fragment loads
// feed 8 WMMAs (0.75 loads/wmma) -- B fragments are reused across both M
// halves, A fragments across all four N tiles.
//
// Workspace layout (floats): Q[4096*1024] | K[4096*1024] | S[4096*4096]
//   = 25,165,824 floats = 96 MB.

#define N_TOK 4096
#define DIM   1024

typedef __attribute__((ext_vector_type(2))) float v2f;
typedef __attribute__((ext_vector_type(8))) float v8f;

static __device__ __forceinline__ v8f wmma_f32_k4(v2f a, v2f b, v8f c) {
  // D(16x16,f32) = A(16x4,f32) * B(4x16,f32) + C
  return __builtin_amdgcn_wmma_f32_16x16x4_f32(
      /*neg_a=*/false, a, /*neg_b=*/false, b,
      /*c_mod=*/(short)0, c, /*reuse_a=*/false, /*reuse_b=*/false);
}

// One wave computes a 32x64 strip of C = A * B^T (+bias), contraction over
// kdim. A is [*, kdim] row-major, B is [*, kdim] row-major (so B^T columns
// are B rows -> contiguous float2 fragment loads for both operands).
static __device__ __forceinline__ void gemm_strip32_nt(
    const float* __restrict__ A, const float* __restrict__ B,
    float* __restrict__ C, const float* __restrict__ bias,
    int m0, int n0, int kdim, int ldc) {
  const int lane = threadIdx.x & 31;
  const int r    = lane & 15;   // row within A-frag / col within B-frag
  const int h    = lane >> 4;   // k-half select (ISA 16x4 A layout)

  const float* ap0 = A + (size_t)(m0 + r) * kdim + 2 * h;
  const float* ap1 = ap0 + (size_t)16 * kdim;
  const float* bp  = B + (size_t)(n0 + r) * kdim + 2 * h;

  v8f acc0[4] = {};
  v8f acc1[4] = {};
  #pragma unroll 4
  for (int k0 = 0; k0 < kdim; k0 += 4) {
    v2f a0 = *(const v2f*)(ap0 + k0);
    v2f a1 = *(const v2f*)(ap1 + k0);
    #pragma unroll
    for (int t = 0; t < 4; ++t) {
      v2f b = *(const v2f*)(bp + (size_t)t * 16 * kdim + k0);
      acc0[t] = wmma_f32_k4(a0, b, acc0[t]);
      acc1[t] = wmma_f32_k4(a1, b, acc1[t]);
    }
  }

  // C/D layout: element (M,N): vgpr = M%8, lane = (N%16) + 16*(M/8)
  #pragma unroll
  for (int t = 0; t < 4; ++t) {
    const int col = n0 + t * 16 + r;
    const float bb = bias ? bias[col] : 0.0f;
    #pragma unroll
    for (int v = 0; v < 8; ++v) {
      C[(size_t)(m0 + h * 8 + v) * ldc + col]      = acc0[t][v] + bb;
      C[(size_t)(m0 + 16 + h * 8 + v) * ldc + col] = acc1[t][v] + bb;
    }
  }
}

// Kernel 1: Q = x@Wq^T + bq and K = x@Wk^T + bk, fused (task id selects).
// 4096 wave-tasks (2 matrices * 128 m-blocks * 16 n-strips), 4 waves/block.
__global__ void qk_proj_kernel(const float* __restrict__ x,
                               const float* __restrict__ Wq,
                               const float* __restrict__ bq,
                               const float* __restrict__ Wk,
                               const float* __restrict__ bk,
                               float* __restrict__ Q, float* __restrict__ K) {
  const int task   = blockIdx.x * (blockDim.x >> 5) + (threadIdx.x >> 5);
  const int matsel = task >> 11;      // 0 -> Q, 1 -> K
  const int s      = task & 2047;
  const int m0     = (s >> 4) * 32;
  const int n0     = (s & 15) * 64;
  const float* W  = matsel ? Wk : Wq;
  const float* bi = matsel ? bk : bq;
  float* C        = matsel ? K : Q;
  gemm_strip32_nt(x, W, C, bi, m0, n0, DIM, DIM);
}

// Kernel 2: S = Q @ K^T  (4096x4096). 8192 wave-tasks, 4 waves/block.
__global__ void scores_kernel(const float* __restrict__ Q,
                              const float* __restrict__ K,
                              float* __restrict__ S) {
  const int task = blockIdx.x * (blockDim.x >> 5) + (threadIdx.x >> 5);
  const int m0   = (task >> 6) * 32;
  const int n0   = (task & 63) * 64;
  gemm_strip32_nt(Q, K, S, nullptr, m0, n0, DIM, N_TOK);
}

// Kernel 3: in-place row softmax over S. One 256-thread block per row,
// 16 elements/thread held in registers, LDS tree reductions.
__global__ void softmax_kernel(float* __restrict__ S) {
  __shared__ float red[256];
  const int tid = threadIdx.x;
  float* row = S + (size_t)blockIdx.x * N_TOK;

  float v[16];
  float mx = -3.402823466e38f;
  #pragma unroll
  for (int i = 0; i < 16; ++i) {
    v[i] = row[tid + i * 256];
    mx = fmaxf(mx, v[i]);
  }
  red[tid] = mx;
  __syncthreads();
  for (int s = 128; s > 0; s >>= 1) {
    if (tid < s) red[tid] = fmaxf(red[tid], red[tid + s]);
    __syncthreads();
  }
  mx = red[0];
  __syncthreads();

  float sum = 0.0f;
  #pragma unroll
  for (int i = 0; i < 16; ++i) {
    v[i] = __expf(v[i] - mx);
    sum += v[i];
  }
  red[tid] = sum;
  __syncthreads();
  for (int s = 128; s > 0; s >>= 1) {
    if (tid < s) red[tid] += red[tid + s];
    __syncthreads();
  }
  const float inv = 1.0f / red[0];

  #pragma unroll
  for (int i = 0; i < 16; ++i) row[tid + i * 256] = v[i] * inv;
}

// Kernel 4: out = P @ x (contraction over 4096 keys). P rows are contiguous
// along K; x is accessed NN-style (two b32 loads per B-fragment, coalesced
// across the 16-lane half). The 32-row M block amortizes each (expensive)
// B fragment over two WMMAs. 2048 wave-tasks, 4 waves/block.
__global__ void attn_out_kernel(const float* __restrict__ P,
                                const float* __restrict__ x,
                                float* __restrict__ out) {
  const int task = blockIdx.x * (blockDim.x >> 5) + (threadIdx.x >> 5);
  const int m0   = (task >> 4) * 32;
  const int d0   = (task & 15) * 64;
  const int lane = threadIdx.x & 31;
  const int r    = lane & 15;
  const int h    = lane >> 4;

  const float* ap0 = P + (size_t)(m0 + r) * N_TOK + 2 * h;
  const float* ap1 = ap0 + (size_t)16 * N_TOK;

  v8f acc0[4] = {};
  v8f acc1[4] = {};
  #pragma unroll 2
  for (int k0 = 0; k0 < N_TOK; k0 += 4) {
    v2f a0 = *(const v2f*)(ap0 + k0);
    v2f a1 = *(const v2f*)(ap1 + k0);
    const float* xr = x + (size_t)(k0 + 2 * h) * DIM + d0 + r;
    #pragma unroll
    for (int t = 0; t < 4; ++t) {
      v2f b;
      b.x = xr[t * 16];         // x[k0+2h    ][d0 + t*16 + r]
      b.y = xr[DIM + t * 16];   // x[k0+2h + 1][d0 + t*16 + r]
      acc0[t] = wmma_f32_k4(a0, b, acc0[t]);
      acc1[t] = wmma_f32_k4(a1, b, acc1[t]);
    }
  }

  #pragma unroll
  for (int t = 0; t < 4; ++t) {
    const int col = d0 + t * 16 + r;
    #pragma unroll
    for (int v = 0; v < 8; ++v) {
      out[(size_t)(m0 + h * 8 + v) * DIM + col]      = acc0[t][v];
      out[(size_t)(m0 + 16 + h * 8 + v) * DIM + col] = acc1[t][v];
    }
  }
}

extern "C" void kernel_launch(void* const* d_in, const int* in_sizes, int n_in,
                              void* d_out, int out_size, void* d_ws,
                              size_t ws_size, hipStream_t stream) {
  (void)in_sizes; (void)n_in; (void)out_size; (void)ws_size;
  const float* x  = (const float*)d_in[0];
  const float* Wq = (const float*)d_in[1];
  const float* bq = (const float*)d_in[2];
  const float* Wk = (const float*)d_in[3];
  const float* bk = (const float*)d_in[4];
  float* out = (float*)d_out;

  float* Q = (float*)d_ws;                      // 4096*1024
  float* K = Q + (size_t)N_TOK * DIM;           // 4096*1024
  float* S = K + (size_t)N_TOK * DIM;           // 4096*4096

  // 4096 wave-tasks / 4 waves per 128-thread block
  qk_proj_kernel<<<1024, 128, 0, stream>>>(x, Wq, bq, Wk, bk, Q, K);
  // 8192 wave-tasks
  scores_kernel<<<2048, 128, 0, stream>>>(Q, K, S);
  // one block per row
  softmax_kernel<<<4096, 256, 0, stream>>>(S);
  // 2048 wave-tasks
  attn_out_kernel<<<512, 128, 0, stream>>>(S, x, out);
}